// I_MLP_34282428957033
// MI455X (gfx1250) — compile-verified
//
#include <hip/hip_runtime.h>

#define NSTEPS 10
#define DT_C 0.1f
#define WPB 4   // waves per block (block = 128 threads)

typedef float v2f __attribute__((ext_vector_type(2)));
typedef float v8f __attribute__((ext_vector_type(8)));

#if __has_builtin(__builtin_amdgcn_tanhf)
#define TANHF(x) __builtin_amdgcn_tanhf(x)
#elif __has_builtin(__builtin_amdgcn_tanh_f32)
#define TANHF(x) __builtin_amdgcn_tanh_f32(x)
#else
#define TANHF(x) tanhf(x)
#endif

static __device__ __forceinline__ v8f wmma4(v2f a, v2f b, v8f c) {
  // V_WMMA_F32_16X16X4_F32: D(16x16,f32) = A(16x4) * B(4x16) + C
  return __builtin_amdgcn_wmma_f32_16x16x4_f32(false, a, false, b, (short)0, c,
                                               false, false);
}

__global__ __launch_bounds__(WPB * 32) void rollout_kernel(
    const float* __restrict__ s_star, const float* __restrict__ s0,
    const float* __restrict__ fc1_w, const float* __restrict__ fc1_b,
    const float* __restrict__ fc2_w, const float* __restrict__ fc2_b,
    const float* __restrict__ rc1_w, const float* __restrict__ rc1_b,
    const float* __restrict__ rc2_w, const float* __restrict__ rc2_b,
    const float* __restrict__ rc3_w, const float* __restrict__ rc3_b,
    const float* __restrict__ rr1_w, const float* __restrict__ rr1_b,
    const float* __restrict__ rr2_w, const float* __restrict__ rr2_b,
    const float* __restrict__ rr3_w, const float* __restrict__ rr3_b,
    float* __restrict__ out, int nrows)
{
  __shared__ __align__(16) float lds[WPB][4][512];

  const int tid = (int)threadIdx.x;
  const int w = tid >> 5;       // wave in block
  const int l = tid & 31;       // lane
  const int n = l & 15;         // WMMA N / M-within-half
  const int kh = l >> 4;        // lane half

  float* stage = &lds[w][0][0]; // C staging (preamble) then L3-D grid
  float* gH    = &lds[w][1][0]; // h1 (cols 0-7 rc, 8-15 rr)
  float* gRC   = &lds[w][2][0]; // h2 rc
  float* gRR   = &lds[w][3][0]; // h2 rr

  const int dsoff = kh * 128 + n;      // C/D layout: + r*16 (+ t*256)
  const int aoff  = n * 16 + 2 * kh;   // A layout:   + k0 + v (+ t*256)

  int row = blockIdx.x * (WPB * 32) + tid;   // this lane's batch row
  const bool valid = (row < nrows);
  if (row >= nrows) row = nrows - 1;

  // ---- uniform folded matrices (scalar loads; tiny) ----
  float Ms[2][2], Mt[2][2], cah[2];
#pragma unroll
  for (int i = 0; i < 2; ++i) {
    float a0 = fc2_w[i*4+0], a1 = fc2_w[i*4+1], a2 = fc2_w[i*4+2], a3 = fc2_w[i*4+3];
#pragma unroll
    for (int k = 0; k < 2; ++k) {
      Ms[i][k] = a0*fc1_w[0*4+k]   + a1*fc1_w[1*4+k]   + a2*fc1_w[2*4+k]   + a3*fc1_w[3*4+k];
      Mt[i][k] = a0*fc1_w[0*4+2+k] + a1*fc1_w[1*4+2+k] + a2*fc1_w[2*4+2+k] + a3*fc1_w[3*4+2+k];
    }
    cah[i] = a0*fc1_b[0] + a1*fc1_b[1] + a2*fc1_b[2] + a3*fc1_b[3] + fc2_b[i];
  }
  const float b3rc0 = rc3_b[0], b3rc1 = rc3_b[1];
  const float db0 = rr3_b[0] - rc3_b[0], db1 = rr3_b[1] - rc3_b[1];

  // ---- per-row state ----
  const float ssx = s_star[row*2+0], ssy = s_star[row*2+1];
  float sx = s0[row*2+0], sy = s0[row*2+1];
  const float g0 = Mt[0][0]*ssx + Mt[0][1]*ssy + cah[0];
  const float g1 = Mt[1][0]*ssx + Mt[1][1]*ssy + cah[1];
  const float ahs0 = (ssx > 0.5f) ? -1.0f : 1.0f;
  const float fpx = ahs0 - g0;       // ah_star - g  (ah err const part)
  const float fpy = 0.0f - g1;

  // ---- stage layer-1 per-row constants, load into C layout (resident) ----
#pragma unroll
  for (int j = 0; j < 8; ++j) {
    stage[l*16 + j]     = rc1_b[j] + rc1_w[j*4+2]*g0 + rc1_w[j*4+3]*g1;
    stage[l*16 + 8 + j] = rr1_b[j] + rr1_w[j*4+2]*g0 + rr1_w[j*4+3]*g1;
  }
  __syncthreads();
  v8f C1[2];
#pragma unroll
  for (int t = 0; t < 2; ++t)
#pragma unroll
    for (int r = 0; r < 8; ++r)
      C1[t][r] = stage[t*256 + r*16 + dsoff];

  // ---- resident B-layout weight operands ----
  const float* w1p = (n < 8) ? rc1_w : rr1_w;
  const int jj = n & 7;
  v2f B1;
#pragma unroll
  for (int v = 0; v < 2; ++v) {
    int k = v + 2*kh;
    float val = 0.0f;
    if (k < 2)
      val = w1p[jj*4+k] + w1p[jj*4+2]*Ms[0][k] + w1p[jj*4+3]*Ms[1][k];
    B1[v] = val;
  }
  v2f B2rc[2], B2rr[2];
#pragma unroll
  for (int c = 0; c < 2; ++c)
#pragma unroll
    for (int v = 0; v < 2; ++v) {
      int k = c*4 + v + 2*kh;                 // k in 0..7
      float vc = rc2_w[(n & 7)*8 + k];
      float vr = rr2_w[(n & 7)*8 + k];
      B2rc[c][v] = (n < 8) ? vc : 0.0f;
      B2rr[c][v] = (n < 8) ? vr : 0.0f;
    }
  v2f B3[4];
#pragma unroll
  for (int c = 0; c < 4; ++c)
#pragma unroll
    for (int v = 0; v < 2; ++v) {
      int k = c*4 + v + 2*kh;                 // logical k in 0..15
      int nr = n & 1;
      float vrc = rc3_w[nr*8 + (k & 7)];
      float vrr = rr3_w[nr*8 + (k & 7)];
      float val = 0.0f;
      if (n < 2)      val = (k < 8) ? vrc : 0.0f;          // ar columns
      else if (n < 4) val = (k < 8) ? -vrc : vrr;          // (ar_curr - ar) columns
      B3[c][v] = val;
    }
  const float b2vrc = (n < 8) ? rc2_b[n & 7] : 0.0f;
  const float b2vrr = (n < 8) ? rr2_b[n & 7] : 0.0f;

  const v8f z8 = {0.f,0.f,0.f,0.f,0.f,0.f,0.f,0.f};

  // ---- rollout ----
  float err = 0.0f;
#pragma unroll 1
  for (int step = 0; step < NSTEPS; ++step) {
    // A for layer 1: tile0 = this half's s, tile1 = other half's s (shfl_xor 16)
    const float sxo = __shfl_xor(sx, 16, 32);
    const float syo = __shfl_xor(sy, 16, 32);
    const bool lo = (l < 16);
    v2f A1[2];
    A1[0][0] = lo ? sx  : 0.0f;  A1[0][1] = lo ? sy  : 0.0f;
    A1[1][0] = lo ? sxo : 0.0f;  A1[1][1] = lo ? syo : 0.0f;

    // ah-error uses s BEFORE the update
    const float ahdx = fpx - (Ms[0][0]*sx + Ms[0][1]*sy);
    const float ahdy = fpy - (Ms[1][0]*sx + Ms[1][1]*sy);

    // layer 1 (both policies in one WMMA per tile), tanh, stash to LDS
#pragma unroll
    for (int t = 0; t < 2; ++t) {
      v8f d1 = wmma4(A1[t], B1, C1[t]);
#pragma unroll
      for (int r = 0; r < 8; ++r)
        gH[t*256 + r*16 + dsoff] = TANHF(d1[r]);
    }

    // layer 2 per policy: K=8 via two chained K=4 WMMAs
#pragma unroll
    for (int t = 0; t < 2; ++t) {
      const int tb = t*256;
      v2f aRC0 = { gH[tb + aoff + 0],  gH[tb + aoff + 1]  };
      v2f aRC1 = { gH[tb + aoff + 4],  gH[tb + aoff + 5]  };
      v2f aRR0 = { gH[tb + aoff + 8],  gH[tb + aoff + 9]  };
      v2f aRR1 = { gH[tb + aoff + 12], gH[tb + aoff + 13] };
      v8f dRC = wmma4(aRC1, B2rc[1], wmma4(aRC0, B2rc[0], z8));
      v8f dRR = wmma4(aRR1, B2rr[1], wmma4(aRR0, B2rr[0], z8));
#pragma unroll
      for (int r = 0; r < 8; ++r) {
        gRC[tb + r*16 + dsoff] = TANHF(dRC[r] + b2vrc);
        gRR[tb + r*16 + dsoff] = TANHF(dRR[r] + b2vrr);
      }
    }

    // layer 3: K=16 over [h2_rc | h2_rr]; cols0-1 = ar, cols2-3 = ar_curr - ar
#pragma unroll
    for (int t = 0; t < 2; ++t) {
      const int tb = t*256;
      v2f a0 = { gRC[tb + aoff + 0], gRC[tb + aoff + 1] };
      v2f a1 = { gRC[tb + aoff + 4], gRC[tb + aoff + 5] };
      v2f a2 = { gRR[tb + aoff + 0], gRR[tb + aoff + 1] };
      v2f a3 = { gRR[tb + aoff + 4], gRR[tb + aoff + 5] };
      v8f d3 = wmma4(a3, B3[3], wmma4(a2, B3[2], wmma4(a1, B3[1], wmma4(a0, B3[0], z8))));
#pragma unroll
      for (int r = 0; r < 8; ++r)
        stage[tb + r*16 + dsoff] = d3[r];
    }

    // per-row extraction: one b128 load per lane (row l, cols 0..3)
    const float4 v = *(const float4*)&stage[l*16];
    const float arx = v.x + b3rc0, ary = v.y + b3rc1;
    const float drx = v.z + db0,  dry = v.w + db1;
    sx += DT_C * arx;
    sy += DT_C * ary;
    const float ex = ssx - sx, ey = ssy - sy;
    err += sqrtf(ex*ex + ey*ey)
         + 0.1f * sqrtf(drx*drx + dry*dry)
         + sqrtf(ahdx*ahdx + ahdy*ahdy);
  }

  if (valid) out[row] = err;
}

extern "C" void kernel_launch(void* const* d_in, const int* in_sizes, int n_in,
                              void* d_out, int out_size, void* d_ws, size_t ws_size,
                              hipStream_t stream) {
  const float* s_star = (const float*)d_in[0];
  const float* s0     = (const float*)d_in[1];
  const float* fc1_w  = (const float*)d_in[2];
  const float* fc1_b  = (const float*)d_in[3];
  const float* fc2_w  = (const float*)d_in[4];
  const float* fc2_b  = (const float*)d_in[5];
  const float* rc1_w  = (const float*)d_in[6];
  const float* rc1_b  = (const float*)d_in[7];
  const float* rc2_w  = (const float*)d_in[8];
  const float* rc2_b  = (const float*)d_in[9];
  const float* rc3_w  = (const float*)d_in[10];
  const float* rc3_b  = (const float*)d_in[11];
  const float* rr1_w  = (const float*)d_in[12];
  const float* rr1_b  = (const float*)d_in[13];
  const float* rr2_w  = (const float*)d_in[14];
  const float* rr2_b  = (const float*)d_in[15];
  const float* rr3_w  = (const float*)d_in[16];
  const float* rr3_b  = (const float*)d_in[17];
  float* out = (float*)d_out;

  const int B = in_sizes[0] / 2;
  const int block = WPB * 32;
  const int grid = (B + block - 1) / block;

  rollout_kernel<<<grid, block, 0, stream>>>(
      s_star, s0, fc1_w, fc1_b, fc2_w, fc2_b,
      rc1_w, rc1_b, rc2_w, rc2_b, rc3_w, rc3_b,
      rr1_w, rr1_b, rr2_w, rr2_b, rr3_w, rr3_b,
      out, B);
}